// Resonator_27565100106270
// MI455X (gfx1250) — compile-verified
//
#include <hip/hip_runtime.h>
#include <stdint.h>

#define NS 24000            // samples per voice
#define SR 48000.0f

typedef unsigned int v4u __attribute__((ext_vector_type(4)));
typedef int          v8i __attribute__((ext_vector_type(8)));
typedef int          v4i __attribute__((ext_vector_type(4)));

#if defined(__has_builtin)
#  if __has_builtin(__builtin_amdgcn_tensor_load_to_lds)
#    define HAVE_TDM 1
#  endif
#  if __has_builtin(__builtin_amdgcn_s_wait_tensorcnt)
#    define HAVE_TENSORCNT 1
#  endif
#  if __has_builtin(__builtin_amdgcn_sched_barrier)
#    define SCHED_FENCE() __builtin_amdgcn_sched_barrier(0)
#  endif
#endif
#ifndef SCHED_FENCE
#  define SCHED_FENCE() __threadfence_block()
#endif

// One voice per workgroup, one wave32 per workgroup.
// LDS holds the full 24000-float buffer (96 KB of the WGP's 320 KB; 3 voices
// fit per WGP). TDM stages the transient row into LDS in one descriptor; the
// constant-lag two-tap comb then runs in place, wave-synchronously.
__global__ __launch_bounds__(32) void resonator_kernel(
    const float* __restrict__ transient,   // [B, NS]
    const float* __restrict__ parameters,  // [B, 3] = {freq, feedback, gain}
    float* __restrict__ out)               // [B, NS]
{
    extern __shared__ float buf[];         // NS floats, offset 0 in LDS

    const int voice = blockIdx.x;
    const int lane  = threadIdx.x;

    const float freq = parameters[voice * 3 + 0];
    const float fb   = parameters[voice * 3 + 1];
    const float gain = parameters[voice * 3 + 2];
    const float D    = SR / freq;          // delay in samples, (24, 960]

    // ---- Stage transient[voice, :] -> LDS via the Tensor Data Mover -----
    const unsigned long long gaddr =
        (unsigned long long)(uintptr_t)(transient + (size_t)voice * NS);
    // Flat shared-aperture pointer: addr[31:0] is the LDS byte offset.
    const unsigned lds_off = (unsigned)(uintptr_t)(void*)buf;

#if defined(HAVE_TDM)
    // D# group 0: count=1 (valid), lds_addr, global_addr[56:0], type=2
    v4u g0;
    g0[0] = 1u;                                        // count=1, user mode
    g0[1] = lds_off;                                   // LDS byte address
    g0[2] = (unsigned)(gaddr & 0xFFFFFFFFull);         // global_addr[31:0]
    g0[3] = (unsigned)((gaddr >> 32) & 0x1FFFFFFull)   // global_addr[56:32]
            | 0x80000000u;                             // type = 2 ("image")
    // D# group 1: data_size=4B, 1-row tile of 24000 elements
    v8i g1;
    g1[0] = (int)(2u << 16);                 // data_size = 2 -> 4 bytes
    g1[1] = (int)((NS & 0xFFFF) << 16);      // tensor_dim0[15:0] in [63:48]
    g1[2] = (int)(1u << 16);                 // tensor_dim0 hi=0 | tensor_dim1=1
    g1[3] = (int)((NS & 0xFFFF) << 16);      // tile_dim0 = 24000
    g1[4] = 1;                               // tile_dim1 = 1, tile_dim2 = 0
    g1[5] = NS;                              // tensor_dim0_stride[31:0]
    g1[6] = 0;                               // stride0 hi | stride1 lo
    g1[7] = 0;
    v4i gz  = {0, 0, 0, 0};                  // groups 2/3 unused (<=2D tensor)
    v8i gz8 = {0, 0, 0, 0, 0, 0, 0, 0};      // 6-arg (clang-23) extra group

    __builtin_amdgcn_tensor_load_to_lds(g0, g1, gz, gz, gz8, 0);
#  if defined(HAVE_TENSORCNT)
    __builtin_amdgcn_s_wait_tensorcnt(0);
#  endif
#else
    // Fallback staging path (no TDM builtin on this toolchain)
    for (int i = lane; i < NS; i += 32)
        buf[i] = transient[(size_t)voice * NS + i];
#endif
    __syncthreads();   // once, before the recurrence

    // ---- In-place two-tap comb, wave-synchronous chunks -----------------
    // Safe parallel window: dist = floor(D)-1 >= 23 (f32 rounding of
    // delayed=(float)i-D keeps every read index strictly below the chunk
    // base; see ulp bound ~0.002 at i<=24000). Low-frequency voices get
    // up to 8 samples per lane per step (ILP hides LDS latency); the
    // worst-case serial chain is set by freq->2000 voices at dist=23.
    const int dist = (int)floorf(D) - 1;
    int chunk;
    if (dist >= 32) {
        chunk = (dist / 32) * 32;
        if (chunk > 256) chunk = 256;      // up to 8 samples per lane
    } else {
        chunk = (dist < 1) ? 1 : dist;     // 23..31: single masked pass
    }

    float* __restrict__ dst = out + (size_t)voice * NS;

    for (int base = 0; base < NS; base += chunk) {
        // All samples in [base, base+chunk) depend only on indices < base.
        for (int off = lane; off < chunk; off += 32) {
            const int i = base + off;
            if (i < NS) {
                float v = buf[i];                   // transient value
                const float delayed = (float)i - D; // faithful per-i compute
                if (delayed >= 0.0f) {
                    const float flf = floorf(delayed);
                    const int   fl  = (int)flf;     // >= 0 here
                    const int   ce  = fl + 1;       // < base (proved above)
                    const float fr  = delayed - flf;
                    const float interp =
                        (1.0f - fr) * buf[fl] + fr * buf[ce];
                    v += fb * interp;
                    buf[i] = v;                     // in-place feedback write
                }
                dst[i] = gain * v;                  // streamed, coalesced
            }
        }
        // Same-wave LDS ops are in-order in hardware; just stop the
        // scheduler from moving anything across the chunk boundary.
        SCHED_FENCE();
    }
}

extern "C" void kernel_launch(void* const* d_in, const int* in_sizes, int n_in,
                              void* d_out, int out_size, void* d_ws, size_t ws_size,
                              hipStream_t stream) {
    const float* transient  = (const float*)d_in[0];   // [B, 24000] f32
    const float* parameters = (const float*)d_in[1];   // [B, 3] f32
    float*       out        = (float*)d_out;           // [B, 24000] f32

    const int B = in_sizes[0] / NS;                    // 512 voices
    const size_t lds_bytes = (size_t)NS * sizeof(float);  // 96 KB / voice

    // Allow >64KB dynamic LDS (CDNA5 WGP has 320 KB). Idempotent, capture-safe.
    (void)hipFuncSetAttribute((const void*)resonator_kernel,
                              hipFuncAttributeMaxDynamicSharedMemorySize,
                              (int)lds_bytes);

    hipLaunchKernelGGL(resonator_kernel, dim3(B), dim3(32), lds_bytes, stream,
                       transient, parameters, out);
}